// Net_28484223107413
// MI455X (gfx1250) — compile-verified
//
#include <hip/hip_runtime.h>
#include <hip/hip_bf16.h>
#include <math.h>

typedef __attribute__((ext_vector_type(2))) float v2f;
typedef __attribute__((ext_vector_type(8))) float v8f;

#define N_NODES 100000
#define N_EDGES 1600000
#define KDIM 128
#define APITCH 132   // A-tile LDS row pitch (floats): 16B-aligned rows, spreads banks

// ---------------------------------------------------------------------------
// Degree accumulation: out_deg[src[e]] += 1 ; in_deg[dst[e]] += 1
// ---------------------------------------------------------------------------
__global__ void degree_kernel(const int* __restrict__ src, const int* __restrict__ dst,
                              float* __restrict__ out_deg, float* __restrict__ in_deg,
                              int n_edges) {
    int e = blockIdx.x * blockDim.x + threadIdx.x;
    if (e < n_edges) {
        atomicAdd(&out_deg[src[e]], 1.0f);
        atomicAdd(&in_deg[dst[e]], 1.0f);
    }
}

// deg -> clip(deg,1)^-0.5   (in place, over 2*N_NODES contiguous floats)
__global__ void norm_kernel(float* __restrict__ deg, int n) {
    int i = blockIdx.x * blockDim.x + threadIdx.x;
    if (i < n) {
        deg[i] = rsqrtf(fmaxf(deg[i], 1.0f));
    }
}

// ---------------------------------------------------------------------------
// y[r,:] = x[r,:] * s[r]     (float4 vectorized, dim = 128)
// ---------------------------------------------------------------------------
__global__ void scale_rows_kernel(const float* __restrict__ x, const float* __restrict__ s,
                                  float* __restrict__ y, int m) {
    int t = blockIdx.x * blockDim.x + threadIdx.x;   // m * 32 threads
    int r = t >> 5;
    int c = (t & 31) << 2;
    if (r < m) {
        float sv = s[r];
        const float4 v = *(const float4*)(x + (size_t)r * KDIM + c);
        float4 o;
        o.x = v.x * sv; o.y = v.y * sv; o.z = v.z * sv; o.w = v.w * sv;
        *(float4*)(y + (size_t)r * KDIM + c) = o;
    }
}

// ---------------------------------------------------------------------------
// Edge gather + scatter-add:  agg[dst[e], :] += hs[src[e], :]
// One thread per (edge, 4-float chunk); a wave covers whole rows ->
// coalesced gathers; scatter uses global_atomic_add_f32.
// cshift = log2(dim/4): 5 for dim=128, 4 for dim=64.
// ---------------------------------------------------------------------------
__global__ void edge_agg_kernel(const float* __restrict__ hs,
                                const int* __restrict__ src, const int* __restrict__ dst,
                                float* __restrict__ agg, int n_edges, int dim, int cshift) {
    int t = blockIdx.x * blockDim.x + threadIdx.x;
    int e = t >> cshift;
    int c = (t & ((1 << cshift) - 1)) << 2;
    if (e < n_edges) {
        int s = src[e];
        int d = dst[e];
        const float4 v = *(const float4*)(hs + (size_t)s * dim + c);
        float* p = agg + (size_t)d * dim + c;
        atomicAdd(p + 0, v.x);
        atomicAdd(p + 1, v.y);
        atomicAdd(p + 2, v.z);
        atomicAdd(p + 3, v.w);
    }
}

// ---------------------------------------------------------------------------
// WMMA GEMM:  out[r, :] = act( row_scale[r] * A[r, :] @ W + bias )
//   A: m x 128 (f32), W: 128 x (NTILES*16) (f32)
// Block = 256 threads = 8 waves, 128 rows per block.
//   - A tile (128 x 128) staged in LDS via global_load_async_to_lds_b128
//     (ASYNCcnt path), row pitch APITCH for alignment/bank spread.
//   - W staged in LDS pair-interleaved: (W[k][n],W[k+1][n]) contiguous so a
//     B fragment is one ds_load_b64 (no packing moves).
//   - V_WMMA_F32_16X16X4_F32 accumulates in fp32 (GEMM is not the bottleneck;
//     precision is free).
// EXEC is all-ones from the barrier to the guarded store (WMMA requirement).
// ---------------------------------------------------------------------------
template <int NTILES>
__global__ void gemm_wmma_kernel(const float* __restrict__ A, const float* __restrict__ W,
                                 const float* __restrict__ bias,
                                 const float* __restrict__ row_scale,
                                 float* __restrict__ out, int m, int relu) {
    constexpr int NCOLS = NTILES * 16;
    extern __shared__ float smem[];
    float* As = smem;                  // 128 * APITCH floats
    float* Wp = smem + 128 * APITCH;   // KDIM * NCOLS floats (pair-interleaved)

    // ---- stage A tile: 128 rows x 128 cols, async copy global -> LDS ----
    {
        const int row0 = blockIdx.x * 128;
        // 4096 float4 chunks / 256 threads = 16 per thread; consecutive
        // threads -> consecutive 16B chunks (fully coalesced).
        for (int i = threadIdx.x; i < 128 * 32; i += blockDim.x) {
            const int row = i >> 5;          // local row 0..127
            const int c4  = (i & 31) << 2;   // float col, multiple of 4
            float* lp = As + row * APITCH + c4;
            const int grow = row0 + row;
            if (grow < m) {
                const float* gp = A + (size_t)grow * KDIM + c4;
                unsigned lds_off = (unsigned)(size_t)(const void*)lp;
                asm volatile("global_load_async_to_lds_b128 %0, %1, off"
                             :: "v"(lds_off), "v"(gp) : "memory");
            } else {
                // zero-fill out-of-range rows (last block only)
                float4 z = {0.0f, 0.0f, 0.0f, 0.0f};
                *(float4*)lp = z;
            }
        }
    }

    // ---- stage W pair-interleaved: Wp[(k/2)*NCOLS + n][0..1] = W[k..k+1][n]
    for (int i = threadIdx.x; i < (KDIM / 2) * NCOLS; i += blockDim.x) {
        const int kp = i / NCOLS;
        const int n  = i - kp * NCOLS;
        float2 w;
        w.x = W[(2 * kp) * NCOLS + n];
        w.y = W[(2 * kp + 1) * NCOLS + n];
        *(float2*)(Wp + ((size_t)kp * NCOLS + n) * 2) = w;
    }

    asm volatile("s_wait_asynccnt 0x0" ::: "memory");
    __syncthreads();

    const int wave = threadIdx.x >> 5;
    const int lane = threadIdx.x & 31;
    const int row0 = blockIdx.x * 128 + wave * 16;

    // A fragment (16x4 f32 layout): lanes 0-15: M=lane,   K=kstep+{0,1}
    //                               lanes16-31: M=lane-16, K=kstep+{2,3}
    const int mloc = lane & 15;
    const int kgrp = (lane >> 4) * 2;
    int rowA = row0 + mloc;
    if (rowA > m - 1) rowA = m - 1;                 // clamped read, EXEC stays full

    const float s = row_scale ? row_scale[rowA] : 1.0f;
    const float* arow = As + (wave * 16 + mloc) * APITCH;

    v8f acc[NTILES];
#pragma unroll
    for (int t = 0; t < NTILES; ++t) acc[t] = (v8f)0.0f;

    for (int k = 0; k < KDIM; k += 4) {
        const int kb = k + kgrp;
        v2f a = *(const v2f*)(arow + kb);            // ds_load_b64
        a.x *= s;
        a.y *= s;
        const float* bbase = Wp + ((size_t)(kb >> 1) * NCOLS) * 2 + mloc * 2;
#pragma unroll
        for (int t = 0; t < NTILES; ++t) {
            const v2f b = *(const v2f*)(bbase + t * 32);   // ds_load_b64
            acc[t] = __builtin_amdgcn_wmma_f32_16x16x4_f32(
                false, a, false, b, (short)0, acc[t], false, false);
        }
    }

    // C/D layout: VGPR i -> lanes 0-15: M=i, lanes 16-31: M=8+i; N = lane&15
    const int rbase = row0 + (lane >> 4) * 8;
#pragma unroll
    for (int t = 0; t < NTILES; ++t) {
        const int col = t * 16 + mloc;
        const float bv = bias ? bias[col] : 0.0f;
#pragma unroll
        for (int i = 0; i < 8; ++i) {
            const int r = rbase + i;
            if (r < m) {
                float v = acc[t][i] + bv;
                if (relu) v = fmaxf(v, 0.0f);
                out[(size_t)r * NCOLS + col] = v;
            }
        }
    }
}

// ---------------------------------------------------------------------------
// Final layer tail: v = relu(norm_dst[r] * agg[r,c] + b2[c]); log_softmax row.
// One wave per row, 2 columns per lane (64 outputs), wave32 shfl reductions.
// ---------------------------------------------------------------------------
__global__ void finalize_kernel(const float* __restrict__ agg, const float* __restrict__ b2,
                                const float* __restrict__ norm_dst,
                                float* __restrict__ out, int m) {
    const int wave = threadIdx.x >> 5;
    const int lane = threadIdx.x & 31;
    const int r = blockIdx.x * 8 + wave;
    if (r >= m) return;

    const float nd = norm_dst[r];
    const float* row = agg + (size_t)r * 64;
    float v0 = fmaxf(nd * row[lane] + b2[lane], 0.0f);
    float v1 = fmaxf(nd * row[lane + 32] + b2[lane + 32], 0.0f);

    float mx = fmaxf(v0, v1);
#pragma unroll
    for (int off = 16; off > 0; off >>= 1)
        mx = fmaxf(mx, __shfl_xor(mx, off, 32));

    float sum = __expf(v0 - mx) + __expf(v1 - mx);
#pragma unroll
    for (int off = 16; off > 0; off >>= 1)
        sum += __shfl_xor(sum, off, 32);

    const float lse = mx + __logf(sum);
    out[(size_t)r * 64 + lane]      = v0 - lse;
    out[(size_t)r * 64 + lane + 32] = v1 - lse;
}

// ---------------------------------------------------------------------------
// Host orchestration
// ---------------------------------------------------------------------------
extern "C" void kernel_launch(void* const* d_in, const int* in_sizes, int n_in,
                              void* d_out, int out_size, void* d_ws, size_t ws_size,
                              hipStream_t stream) {
    const float* feats = (const float*)d_in[0];
    const float* W0    = (const float*)d_in[1];
    const float* b0    = (const float*)d_in[2];
    const float* W1    = (const float*)d_in[3];
    const float* b1    = (const float*)d_in[4];
    const float* W2    = (const float*)d_in[5];
    const float* b2    = (const float*)d_in[6];
    const int*   src   = (const int*)d_in[7];
    const int*   dst   = (const int*)d_in[8];
    float* out = (float*)d_out;

    const int M = N_NODES;
    const int E = N_EDGES;

    float* ws = (float*)d_ws;
    float* norm_src = ws;                       // M floats (out_deg -> norm)
    float* norm_dst = ws + M;                   // M floats (in_deg  -> norm)
    float* bufX = ws + 2 * (size_t)M;           // M*128 floats
    float* bufY = bufX + (size_t)M * KDIM;      // M*128 floats

    const size_t rowBytes128 = (size_t)M * KDIM * sizeof(float);
    const size_t lds128 = (128 * APITCH + 128 * 128) * sizeof(float);
    const size_t lds64  = (128 * APITCH + 128 * 64) * sizeof(float);

    // --- degrees -> norms ---
    hipMemsetAsync(norm_src, 0, 2 * (size_t)M * sizeof(float), stream);
    degree_kernel<<<(E + 255) / 256, 256, 0, stream>>>(src, dst, norm_src, norm_dst, E);
    norm_kernel<<<(2 * M + 255) / 256, 256, 0, stream>>>(norm_src, 2 * M);

    const int gemmBlocks = (M + 127) / 128;
    const int rowScaleBlocks = (M * 32 + 255) / 256;
    const int agg128Blocks = (int)(((size_t)E * 32 + 255) / 256);
    const int agg64Blocks  = (int)(((size_t)E * 16 + 255) / 256);

    // --- layer 0 : h = relu(norm_dst * Agg(norm_src * feats) @ W0 + b0) ---
    scale_rows_kernel<<<rowScaleBlocks, 256, 0, stream>>>(feats, norm_src, bufY, M);
    hipMemsetAsync(bufX, 0, rowBytes128, stream);
    edge_agg_kernel<<<agg128Blocks, 256, 0, stream>>>(bufY, src, dst, bufX, E, KDIM, 5);
    gemm_wmma_kernel<8><<<gemmBlocks, 256, lds128, stream>>>(
        bufX, W0, b0, norm_dst, bufY, M, 1);

    // --- layer 1 ---
    scale_rows_kernel<<<rowScaleBlocks, 256, 0, stream>>>(bufY, norm_src, bufX, M);
    hipMemsetAsync(bufY, 0, rowBytes128, stream);
    edge_agg_kernel<<<agg128Blocks, 256, 0, stream>>>(bufX, src, dst, bufY, E, KDIM, 5);
    gemm_wmma_kernel<8><<<gemmBlocks, 256, lds128, stream>>>(
        bufY, W1, b1, norm_dst, bufX, M, 1);

    // --- layer 2 : exploit linearity, GEMM (128->64) BEFORE aggregation to
    //     halve edge traffic:  Agg(x) @ W2 == Agg(x @ W2) ---
    scale_rows_kernel<<<rowScaleBlocks, 256, 0, stream>>>(bufX, norm_src, bufY, M);
    gemm_wmma_kernel<4><<<gemmBlocks, 256, lds64, stream>>>(
        bufY, W2, nullptr, nullptr, bufX, M, 0);          // bufX: M x 64
    hipMemsetAsync(bufY, 0, (size_t)M * 64 * sizeof(float), stream);
    edge_agg_kernel<<<agg64Blocks, 256, 0, stream>>>(bufX, src, dst, bufY, E, 64, 4);
    finalize_kernel<<<(M + 7) / 8, 256, 0, stream>>>(bufY, b2, norm_dst, out, M);
}